// ImprovedFastDCTMVTracker_4818953306517
// MI455X (gfx1250) — compile-verified
//
#include <hip/hip_runtime.h>
#include <math.h>

#define NBOX 4096
#define GG   3600
#define DIN  96
#define HDIM 128

typedef __attribute__((ext_vector_type(2))) float v2f;
typedef __attribute__((ext_vector_type(8))) float v8f;

// ---------------------------------------------------------------------------
// Kernel 1: motion stats + featurizer. One wave (32 lanes) per box, 8 waves
// per block. Motion grid (2x60x60 = 28.8KB) cached in LDS.
// Produces x = concat(mv_feat, box_feat) : N x 96 into workspace.
// ---------------------------------------------------------------------------
__global__ __launch_bounds__(256)
void k_features(const float* __restrict__ mv, const float* __restrict__ boxes,
                const float* __restrict__ W_stats, const float* __restrict__ b_stats,
                const float* __restrict__ g_stats, const float* __restrict__ be_stats,
                const float* __restrict__ W_b1, const float* __restrict__ b_b1,
                const float* __restrict__ g_b1, const float* __restrict__ be_b1,
                const float* __restrict__ W_b2, const float* __restrict__ b_b2,
                const float* __restrict__ g_b2, const float* __restrict__ be_b2,
                float* __restrict__ xout)
{
    __shared__ float smv[2 * GG];
    const int tid = threadIdx.x;
    for (int i = tid; i < 2 * GG; i += 256) smv[i] = mv[i];
    __syncthreads();

    const int lane = tid & 31;
    const int wave = tid >> 5;
    const int b    = blockIdx.x * 8 + wave;

    const float bx0 = boxes[b * 4 + 0], bx1 = boxes[b * 4 + 1];
    const float bx2 = boxes[b * 4 + 2], bx3 = boxes[b * 4 + 3];
    const float inv = 1.0f / 960.0f;
    const float bn0 = fminf(fmaxf(bx0 * inv, 0.f), 1.f);
    const float bn1 = fminf(fmaxf(bx1 * inv, 0.f), 1.f);
    const float bn2 = fminf(fmaxf(bx2 * inv, 0.f), 1.f);
    const float bn3 = fminf(fmaxf(bx3 * inv, 0.f), 1.f);

    const int gx1 = min(max((int)floorf(bn0 * 60.f), 0), 59);
    const int gy1 = min(max((int)floorf(bn1 * 60.f), 0), 59);
    const int gx2 = min(max((int)ceilf(bn2 * 60.f), gx1 + 1), 60);
    const int gy2 = min(max((int)ceilf(bn3 * 60.f), gy1 + 1), 60);
    const int rw = gx2 - gx1, rh = gy2 - gy1, cnt = rw * rh;

    float s0 = 0.f, s1 = 0.f, q0 = 0.f, q1 = 0.f;
    float hi0 = -1e30f, hi1 = -1e30f, lo0 = 1e30f, lo1 = 1e30f;
    for (int i = lane; i < cnt; i += 32) {
        int r = i / rw, c = i - r * rw;
        int cell = (gy1 + r) * 60 + (gx1 + c);
        float v0 = smv[cell], v1 = smv[GG + cell];
        s0 += v0; q0 += v0 * v0; hi0 = fmaxf(hi0, v0); lo0 = fminf(lo0, v0);
        s1 += v1; q1 += v1 * v1; hi1 = fmaxf(hi1, v1); lo1 = fminf(lo1, v1);
    }
    for (int m = 16; m >= 1; m >>= 1) {
        s0 += __shfl_xor(s0, m, 32); q0 += __shfl_xor(q0, m, 32);
        s1 += __shfl_xor(s1, m, 32); q1 += __shfl_xor(q1, m, 32);
        hi0 = fmaxf(hi0, __shfl_xor(hi0, m, 32));
        lo0 = fminf(lo0, __shfl_xor(lo0, m, 32));
        hi1 = fmaxf(hi1, __shfl_xor(hi1, m, 32));
        lo1 = fminf(lo1, __shfl_xor(lo1, m, 32));
    }
    const float fc = (float)cnt;
    float st[6];
    st[0] = s0 / fc; st[1] = s1 / fc;
    st[2] = sqrtf(fmaxf(q0 / fc - st[0] * st[0], 0.f));
    st[3] = sqrtf(fmaxf(q1 / fc - st[1] * st[1], 0.f));
    st[4] = hi0 - lo0; st[5] = hi1 - lo1;

    // mv_feat = relu(LN(stats @ W_stats.T + b_stats)); lane handles j=lane, j=lane+32
    float ya = b_stats[lane], yb = b_stats[lane + 32];
    #pragma unroll
    for (int k = 0; k < 6; ++k) {
        ya += st[k] * W_stats[lane * 6 + k];
        yb += st[k] * W_stats[(lane + 32) * 6 + k];
    }
    float ss = ya + yb, sq = ya * ya + yb * yb;
    for (int m = 16; m >= 1; m >>= 1) { ss += __shfl_xor(ss, m, 32); sq += __shfl_xor(sq, m, 32); }
    float mu = ss * (1.f / 64.f);
    float var = sq * (1.f / 64.f) - mu * mu;
    float rin = rsqrtf(var + 1e-5f);
    xout[b * DIN + lane]      = fmaxf((ya - mu) * rin * g_stats[lane]      + be_stats[lane], 0.f);
    xout[b * DIN + lane + 32] = fmaxf((yb - mu) * rin * g_stats[lane + 32] + be_stats[lane + 32], 0.f);

    // box branch: 4 -> 32 (LN,relu) -> 32 (LN,relu); lane = feature index
    float t1 = b_b1[lane]
             + bn0 * W_b1[lane * 4 + 0] + bn1 * W_b1[lane * 4 + 1]
             + bn2 * W_b1[lane * 4 + 2] + bn3 * W_b1[lane * 4 + 3];
    float s = t1, q = t1 * t1;
    for (int m = 16; m >= 1; m >>= 1) { s += __shfl_xor(s, m, 32); q += __shfl_xor(q, m, 32); }
    mu = s * (1.f / 32.f); var = q * (1.f / 32.f) - mu * mu; rin = rsqrtf(var + 1e-5f);
    float h1 = fmaxf((t1 - mu) * rin * g_b1[lane] + be_b1[lane], 0.f);

    float t2 = b_b2[lane];
    #pragma unroll 8
    for (int k = 0; k < 32; ++k) t2 += __shfl(h1, k, 32) * W_b2[lane * 32 + k];
    s = t2; q = t2 * t2;
    for (int m = 16; m >= 1; m >>= 1) { s += __shfl_xor(s, m, 32); q += __shfl_xor(q, m, 32); }
    mu = s * (1.f / 32.f); var = q * (1.f / 32.f) - mu * mu; rin = rsqrtf(var + 1e-5f);
    xout[b * DIN + 64 + lane] = fmaxf((t2 - mu) * rin * g_b2[lane] + be_b2[lane], 0.f);
}

// ---------------------------------------------------------------------------
// Kernel 2: gates = x @ W_ih.T + (b_ih+b_hh), fused LSTM (h0=c0=0, f-gate dead).
// 16 boxes per block, 4 waves. Each wave owns 2 h-column tiles; for each it
// runs 3 interleaved WMMA accumulator chains (i, g, o gates) sharing A.
// V_WMMA_F32_16X16X4_F32: full f32 precision.
// ---------------------------------------------------------------------------
__global__ __launch_bounds__(128)
void k_lstm(const float* __restrict__ x, const float* __restrict__ W_ih,
            const float* __restrict__ b_ih, const float* __restrict__ b_hh,
            float* __restrict__ hout)
{
    __shared__ float sx[16 * DIN];
    const int tid  = threadIdx.x;
    const int box0 = blockIdx.x * 16;
    for (int i = tid; i < 16 * DIN; i += 128) sx[i] = x[box0 * DIN + i];
    __syncthreads();

    const int lane = tid & 31;
    const int wave = tid >> 5;
    const int nn   = lane & 15;     // A: M index; B: N index; C: N index
    const int half = lane >> 4;
    const int koff = half * 2;      // K offset within a K=4 step

    const float* Arow = &sx[nn * DIN + koff];

    for (int tt = 0; tt < 2; ++tt) {
        const int t  = wave * 2 + tt;          // h tile 0..7 (16 cols each)
        const int ci = t * 16 + nn;            // i-gate column
        const int cg = 256 + t * 16 + nn;      // g-gate column
        const int co = 384 + t * 16 + nn;      // o-gate column
        const float* Bi = &W_ih[ci * DIN + koff];
        const float* Bg = &W_ih[cg * DIN + koff];
        const float* Bo = &W_ih[co * DIN + koff];
        __builtin_prefetch(Bi, 0, 0);
        __builtin_prefetch(Bg, 0, 0);
        __builtin_prefetch(Bo, 0, 0);

        v8f ai = {}, ag = {}, ao = {};
        #pragma unroll
        for (int k0 = 0; k0 < DIN; k0 += 4) {
            v2f a  = *(const v2f*)(Arow + k0);
            v2f bi = *(const v2f*)(Bi + k0);
            v2f bg = *(const v2f*)(Bg + k0);
            v2f bo = *(const v2f*)(Bo + k0);
            ai = __builtin_amdgcn_wmma_f32_16x16x4_f32(false, a, false, bi, (short)0, ai, false, false);
            ag = __builtin_amdgcn_wmma_f32_16x16x4_f32(false, a, false, bg, (short)0, ag, false, false);
            ao = __builtin_amdgcn_wmma_f32_16x16x4_f32(false, a, false, bo, (short)0, ao, false, false);
        }
        const float bbi = b_ih[ci] + b_hh[ci];
        const float bbg = b_ih[cg] + b_hh[cg];
        const float bbo = b_ih[co] + b_hh[co];
        #pragma unroll
        for (int e = 0; e < 8; ++e) {
            const int m = e + half * 8;        // box row within tile
            float iv = ai[e] + bbi;
            float gv = ag[e] + bbg;
            float ov = ao[e] + bbo;
            float cc = (1.f / (1.f + expf(-iv))) * tanhf(gv);   // c0 == 0
            float hv = (1.f / (1.f + expf(-ov))) * tanhf(cc);
            hout[(box0 + m) * HDIM + t * 16 + nn] = hv;
        }
    }
}

// ---------------------------------------------------------------------------
// Kernel 3: three heads. First layers (128->64, x3 heads = 192 cols) as WMMA
// GEMM, relu into LDS; tiny second layers (64->2/2/1) + box decode.
// ---------------------------------------------------------------------------
__global__ __launch_bounds__(128)
void k_heads(const float* __restrict__ h, const float* __restrict__ boxes,
             const float* __restrict__ W_p1, const float* __restrict__ b_p1,
             const float* __restrict__ W_p2, const float* __restrict__ b_p2,
             const float* __restrict__ W_s1, const float* __restrict__ b_s1,
             const float* __restrict__ W_s2, const float* __restrict__ b_s2,
             const float* __restrict__ W_c1, const float* __restrict__ b_c1,
             const float* __restrict__ W_c2, const float* __restrict__ b_c2,
             float* __restrict__ out)
{
    __shared__ float sh[16 * HDIM];
    __shared__ float su[16 * 192];
    __shared__ float sres[16 * 8];
    const int tid  = threadIdx.x;
    const int box0 = blockIdx.x * 16;
    for (int i = tid; i < 16 * HDIM; i += 128) sh[i] = h[box0 * HDIM + i];
    __syncthreads();

    const int lane = tid & 31;
    const int wave = tid >> 5;
    const int nn   = lane & 15;
    const int half = lane >> 4;
    const int koff = half * 2;
    const float* Arow = &sh[nn * HDIM + koff];

    for (int ti = 0; ti < 3; ++ti) {
        const int ct   = wave * 3 + ti;    // 0..11 (uniform per wave)
        const int head = ct >> 2;          // 0=pos 1=size 2=conf
        const int cih  = (ct & 3) * 16;    // column within head (0..48)
        const float* W  = (head == 0) ? W_p1 : (head == 1) ? W_s1 : W_c1;
        const float* bp = (head == 0) ? b_p1 : (head == 1) ? b_s1 : b_c1;
        const float* Bp = &W[(cih + nn) * HDIM + koff];
        __builtin_prefetch(Bp, 0, 0);

        v8f acc = {};
        #pragma unroll
        for (int k0 = 0; k0 < HDIM; k0 += 4) {
            v2f a  = *(const v2f*)(Arow + k0);
            v2f bb = *(const v2f*)(Bp + k0);
            acc = __builtin_amdgcn_wmma_f32_16x16x4_f32(false, a, false, bb, (short)0, acc, false, false);
        }
        const float bias = bp[cih + nn];
        #pragma unroll
        for (int e = 0; e < 8; ++e) {
            const int m = e + half * 8;
            su[m * 192 + head * 64 + cih + nn] = fmaxf(acc[e] + bias, 0.f);
        }
    }
    __syncthreads();

    if (tid < 80) {
        const int bb   = tid & 15;
        const int unit = tid >> 4;          // 0,1=pos 2,3=size 4=conf
        const float* u = &su[bb * 192];
        const float* W2; float b2v; int off;
        if (unit < 2)      { W2 = W_p2 + unit * 64;       b2v = b_p2[unit];     off = 0;   }
        else if (unit < 4) { W2 = W_s2 + (unit - 2) * 64; b2v = b_s2[unit - 2]; off = 64;  }
        else               { W2 = W_c2;                   b2v = b_c2[0];        off = 128; }
        float acc = b2v;
        #pragma unroll 8
        for (int k = 0; k < 64; ++k) acc += u[off + k] * W2[k];
        sres[bb * 8 + unit] = acc;
    }
    __syncthreads();

    if (tid < 16) {
        const int b = box0 + tid;
        float x1 = boxes[b * 4 + 0], y1 = boxes[b * 4 + 1];
        float x2 = boxes[b * 4 + 2], y2 = boxes[b * 4 + 3];
        float cx = (x1 + x2) * 0.5f, cy = (y1 + y2) * 0.5f;
        float w = x2 - x1, hg = y2 - y1;
        float p0 = sres[tid * 8 + 0], p1 = sres[tid * 8 + 1];
        float s0 = sres[tid * 8 + 2], s1 = sres[tid * 8 + 3];
        float cf = sres[tid * 8 + 4];
        float ncx = cx + p0, ncy = cy + p1;
        float nw = w * expf(s0), nh = hg * expf(s1);
        out[b * 5 + 0] = ncx - nw * 0.5f;
        out[b * 5 + 1] = ncy - nh * 0.5f;
        out[b * 5 + 2] = ncx + nw * 0.5f;
        out[b * 5 + 3] = ncy + nh * 0.5f;
        out[b * 5 + 4] = 1.f / (1.f + expf(-cf));
    }
}

// ---------------------------------------------------------------------------
extern "C" void kernel_launch(void* const* d_in, const int* in_sizes, int n_in,
                              void* d_out, int out_size, void* d_ws, size_t ws_size,
                              hipStream_t stream) {
    const float* mv       = (const float*)d_in[0];
    const float* boxes    = (const float*)d_in[1];
    const float* W_stats  = (const float*)d_in[2];
    const float* b_stats  = (const float*)d_in[3];
    const float* g_stats  = (const float*)d_in[4];
    const float* be_stats = (const float*)d_in[5];
    const float* W_b1     = (const float*)d_in[6];
    const float* b_b1     = (const float*)d_in[7];
    const float* g_b1     = (const float*)d_in[8];
    const float* be_b1    = (const float*)d_in[9];
    const float* W_b2     = (const float*)d_in[10];
    const float* b_b2     = (const float*)d_in[11];
    const float* g_b2     = (const float*)d_in[12];
    const float* be_b2    = (const float*)d_in[13];
    const float* W_ih     = (const float*)d_in[14];
    // d_in[15] = W_hh: unused, h0 == 0
    const float* b_ih     = (const float*)d_in[16];
    const float* b_hh     = (const float*)d_in[17];
    const float* W_p1     = (const float*)d_in[18];
    const float* b_p1     = (const float*)d_in[19];
    const float* W_p2     = (const float*)d_in[20];
    const float* b_p2     = (const float*)d_in[21];
    const float* W_s1     = (const float*)d_in[22];
    const float* b_s1     = (const float*)d_in[23];
    const float* W_s2     = (const float*)d_in[24];
    const float* b_s2     = (const float*)d_in[25];
    const float* W_c1     = (const float*)d_in[26];
    const float* b_c1     = (const float*)d_in[27];
    const float* W_c2     = (const float*)d_in[28];
    const float* b_c2     = (const float*)d_in[29];

    float* xbuf = (float*)d_ws;                 // NBOX x 96
    float* hbuf = xbuf + (size_t)NBOX * DIN;    // NBOX x 128

    k_features<<<NBOX / 8, 256, 0, stream>>>(mv, boxes,
        W_stats, b_stats, g_stats, be_stats,
        W_b1, b_b1, g_b1, be_b1,
        W_b2, b_b2, g_b2, be_b2, xbuf);

    k_lstm<<<NBOX / 16, 128, 0, stream>>>(xbuf, W_ih, b_ih, b_hh, hbuf);

    k_heads<<<NBOX / 16, 128, 0, stream>>>(hbuf, boxes,
        W_p1, b_p1, W_p2, b_p2,
        W_s1, b_s1, W_s2, b_s2,
        W_c1, b_c1, W_c2, b_c2, (float*)d_out);
}